// SelectDecoder_35691178229879
// MI455X (gfx1250) — compile-verified
//
#include <hip/hip_runtime.h>
#include <math.h>

// Problem constants (reference: B=64, N=2048, D=256, H=8, dh=32, S=10)
#define B_   64
#define N_   2048
#define D_   256
#define H_   8
#define HP_  16          // head dim padded to 16 for WMMA tiles
#define DH_  32
#define S_   10
#define CLIPV 10.0f
#define NEGV  -1000000000.0f

typedef __attribute__((ext_vector_type(2))) float v2f;
typedef __attribute__((ext_vector_type(4))) float v4f;
typedef __attribute__((ext_vector_type(8))) float v8f;

// ---------------------------------------------------------------------------
// fp32 WMMA: D(16x16) = A(16x4) * B(4x16) + C.  wave32, all lanes active.
// A layout (ISA 7.12.2): lanes 0-15 -> M=lane, VGPR0=K0, VGPR1=K1;
//                        lanes 16-31 -> M=lane-16, VGPR0=K2, VGPR1=K3.
// C/D layout: lanes 0-15 -> N=lane, VGPR r -> M=r; lanes 16-31 -> N=lane-16, M=8+r.
// ---------------------------------------------------------------------------
__device__ __forceinline__ v8f wmma_f32(v2f a, v2f b, v8f c) {
  return __builtin_amdgcn_wmma_f32_16x16x4_f32(
      /*neg_a=*/false, a, /*neg_b=*/false, b,
      /*c_mod=*/(short)0, c, /*reuse_a=*/false, /*reuse_b=*/false);
}

// ---- init: mask=0, entropy=0, zero padded operand buffers (grid-stride) ----
__global__ void k_init(int* __restrict__ mask, float* __restrict__ ent,
                       float* __restrict__ q2pad, float* __restrict__ apad) {
  int stride = gridDim.x * 256;
  for (int i = blockIdx.x * 256 + threadIdx.x; i < B_ * HP_ * N_; i += stride) {
    apad[i] = 0.f;
    if (i < B_ * HP_ * D_) q2pad[i] = 0.f;
    if (i < B_ * N_)       mask[i]  = 0;
    if (i < B_)            ent[i]   = 0.f;
  }
}

// --------------------------- h_mean = mean_n X ------------------------------
__global__ void k_mean_partial(const float* __restrict__ X, float* __restrict__ hpart) {
  int b = blockIdx.x, c = blockIdx.y, t = threadIdx.x;
  const float* p = X + ((size_t)b * N_ + (size_t)c * 128) * D_ + t;
  float s = 0.f;
  for (int n = 0; n < 128; ++n) s += p[(size_t)n * D_];
  hpart[((size_t)b * 16 + c) * D_ + t] = s;
}
__global__ void k_mean_final(const float* __restrict__ hpart, float* __restrict__ hmean) {
  int b = blockIdx.x, t = threadIdx.x;
  float s = 0.f;
  for (int c = 0; c < 16; ++c) s += hpart[((size_t)b * 16 + c) * D_ + t];
  hmean[b * D_ + t] = s * (1.0f / (float)N_);
}

// --------------------------- q = h_mean @ Wq^T + bq -------------------------
__global__ void k_q(const float* __restrict__ hmean, const float* __restrict__ Wq,
                    const float* __restrict__ bq, float* __restrict__ q) {
  __shared__ float hm[D_];
  int b = blockIdx.x, t = threadIdx.x;
  hm[t] = hmean[b * D_ + t];
  __syncthreads();
  const float* wr = Wq + (size_t)t * D_;
  float s = 0.f;
  for (int d = 0; d < D_; ++d) s += hm[d] * wr[d];
  q[b * D_ + t] = s + bq[t];
}

// -------- Q2pad[b,h,e] = sum_j q[b,h*dh+j] * Wk[h*dh+j, e]  (rows 8..15 = 0) -
__global__ void k_q2(const float* __restrict__ q, const float* __restrict__ Wk,
                     float* __restrict__ q2pad) {
  __shared__ float qh[DH_];
  int b = blockIdx.x, h = blockIdx.y, t = threadIdx.x;
  if (t < DH_) qh[t] = q[b * D_ + h * DH_ + t];
  __syncthreads();
  float s = 0.f;
  for (int j = 0; j < DH_; ++j) s += qh[j] * Wk[(size_t)(h * DH_ + j) * D_ + t];
  q2pad[((size_t)b * HP_ + h) * D_ + t] = s;
}

// ------------- s_mha[b,h,n] = scale * X[b,n,:] . Q2[b,h,:]   (WMMA) ---------
// grid (B, N/128), block 256 = 8 waves; wave handles a 16-row n-tile.
// M = n-tile row, Ncol = head (8 valid of 16, pad rows are zero), K = D.
__global__ void k_smha(const float* __restrict__ X, const float* __restrict__ q2pad,
                       float* __restrict__ smha) {
  int b    = blockIdx.x;
  int wave = threadIdx.x >> 5;
  int lane = threadIdx.x & 31;
  int n0   = (blockIdx.y * 8 + wave) * 16;
  int mrow = lane & 15;
  int khalf = lane >> 4;
  const float* Xrow  = X     + ((size_t)b * N_  + n0 + mrow) * D_ + 2 * khalf;
  const float* Q2row = q2pad + ((size_t)b * HP_ + mrow)      * D_ + 2 * khalf;
  v8f c = {0.f, 0.f, 0.f, 0.f, 0.f, 0.f, 0.f, 0.f};
#pragma unroll 4
  for (int d0 = 0; d0 < D_; d0 += 4) {
    v2f a  = *reinterpret_cast<const v2f*>(Xrow  + d0); // A[M=n, K=dk..dk+1]
    v2f bb = *reinterpret_cast<const v2f*>(Q2row + d0); // B[K=dk..dk+1, N=head]
    c = wmma_f32(a, bb, c);
  }
  const float scale = 0.17677669529663687f;             // 1/sqrt(32)
  if (mrow < H_) {
    int hcol = mrow;
#pragma unroll
    for (int r = 0; r < 8; ++r) {
      int M = r + 8 * khalf;
      smha[((size_t)b * H_ + hcol) * N_ + n0 + M] = c[r] * scale;
    }
  }
}

// ---------- masked softmax over n of s_mha -> apad (padded head rows) -------
__global__ void k_softmax(const float* __restrict__ smha, const int* __restrict__ mask,
                          float* __restrict__ apad) {
  __shared__ float red[256];
  int b = blockIdx.x, h = blockIdx.y, t = threadIdx.x;
  const float* src = smha + ((size_t)b * H_ + h) * N_;
  const int*   mk  = mask + (size_t)b * N_;
  float vals[8];
  float m = -3.0e38f;
#pragma unroll
  for (int j = 0; j < 8; ++j) {
    int n = t + j * 256;
    float v = mk[n] ? NEGV : src[n];
    vals[j] = v;
    m = fmaxf(m, v);
  }
  red[t] = m; __syncthreads();
  for (int s = 128; s > 0; s >>= 1) { if (t < s) red[t] = fmaxf(red[t], red[t + s]); __syncthreads(); }
  m = red[0]; __syncthreads();
  float sum = 0.f;
#pragma unroll
  for (int j = 0; j < 8; ++j) { vals[j] = expf(vals[j] - m); sum += vals[j]; }
  red[t] = sum; __syncthreads();
  for (int s = 128; s > 0; s >>= 1) { if (t < s) red[t] += red[t + s]; __syncthreads(); }
  float inv = 1.0f / red[0];
  float* dst = apad + ((size_t)b * HP_ + h) * N_;
#pragma unroll
  for (int j = 0; j < 8; ++j) dst[t + j * 256] = vals[j] * inv;
}

// --------------- y[b,h,:] = sum_n a[b,h,n] * X[b,n,:]   (WMMA) --------------
// grid (B, D/16), block 256 = 8 waves; wave w accumulates K-range [w*256,+256),
// partial 16x16 tiles reduced through LDS.  M=head(padded), Ncol=d-tile, K=n.
__global__ void k_y(const float* __restrict__ apad, const float* __restrict__ X,
                    float* __restrict__ y) {
  __shared__ float part[8 * 256];
  int b    = blockIdx.x;
  int d0   = blockIdx.y * 16;
  int wave = threadIdx.x >> 5;
  int lane = threadIdx.x & 31;
  int mrow = lane & 15;
  int khalf = lane >> 4;
  const float* Ab = apad + ((size_t)b * HP_ + mrow) * N_ + 2 * khalf;
  const float* Xb = X + (size_t)b * N_ * D_ + d0;
  v8f c = {0.f, 0.f, 0.f, 0.f, 0.f, 0.f, 0.f, 0.f};
  int kbase = wave * 256;
#pragma unroll 4
  for (int k0 = kbase; k0 < kbase + 256; k0 += 4) {
    v2f a = *reinterpret_cast<const v2f*>(Ab + k0);     // A[M=head, K=kr..kr+1]
    int kr = k0 + 2 * khalf;
    v2f bb;
    bb.x = Xb[(size_t)kr * D_ + mrow];                  // B[K=kr,   N=d]
    bb.y = Xb[(size_t)(kr + 1) * D_ + mrow];            // B[K=kr+1, N=d]
    c = wmma_f32(a, bb, c);
  }
#pragma unroll
  for (int r = 0; r < 8; ++r)
    part[wave * 256 + (r + 8 * khalf) * 16 + mrow] = c[r];
  __syncthreads();
  int t = threadIdx.x;
  if (t < H_ * 16) {                                    // t = head*16 + dcol
    float s = 0.f;
    for (int w = 0; w < 8; ++w) s += part[w * 256 + t];
    int h = t >> 4, dc = t & 15;
    y[((size_t)b * H_ + h) * D_ + d0 + dc] = s;
  }
}

// ------- fused small chain: ctx -> attn_out -> qs -> qs2  (per batch) -------
__global__ void k_chain(const float* __restrict__ y,
                        const float* __restrict__ Wv,  const float* __restrict__ bv,
                        const float* __restrict__ Wout, const float* __restrict__ bout,
                        const float* __restrict__ Wsq, const float* __restrict__ Wsk,
                        float* __restrict__ qs2) {
  __shared__ float yl[H_ * D_];
  __shared__ float ctx[D_];
  __shared__ float att[D_];
  __shared__ float qsl[D_];
  int b = blockIdx.x, t = threadIdx.x;
  for (int i = t; i < H_ * D_; i += 256) yl[i] = y[(size_t)b * H_ * D_ + i];
  __syncthreads();
  { // ctx[b, h*dh+d2] = Wv[h*dh+d2,:] . y[b,h,:] + bv
    int h = t >> 5;
    const float* wr = Wv + (size_t)t * D_;
    const float* yh = yl + h * D_;
    float s = 0.f;
    for (int d = 0; d < D_; ++d) s += wr[d] * yh[d];
    ctx[t] = s + bv[t];
  }
  __syncthreads();
  { // attn_out = ctx @ out_w^T + out_b
    const float* wr = Wout + (size_t)t * D_;
    float s = 0.f;
    for (int c = 0; c < D_; ++c) s += wr[c] * ctx[c];
    att[t] = s + bout[t];
  }
  __syncthreads();
  { // qs = attn_out @ sha_wq^T
    const float* wr = Wsq + (size_t)t * D_;
    float s = 0.f;
    for (int c = 0; c < D_; ++c) s += wr[c] * att[c];
    qsl[t] = s;
  }
  __syncthreads();
  { // qs2 = qs @ sha_wk   (so u = X . qs2)
    float s = 0.f;
    for (int j = 0; j < D_; ++j) s += qsl[j] * Wsk[(size_t)j * D_ + t];
    qs2[b * D_ + t] = s;
  }
}

// --------- u[b,n] = CLIP*tanh(scale2 * X[b,n,:] . qs2[b,:]), masked ---------
__global__ void k_u(const float* __restrict__ X, const float* __restrict__ qs2,
                    const int* __restrict__ mask, float* __restrict__ u) {
  __shared__ __align__(16) float ql[D_];
  int b = blockIdx.x, t = threadIdx.x;
  ql[t] = qs2[b * D_ + t];
  __syncthreads();
  int n = blockIdx.y * 256 + t;
  const v4f* row = reinterpret_cast<const v4f*>(X + ((size_t)b * N_ + n) * D_);
  const v4f* q4  = reinterpret_cast<const v4f*>(ql);
  float s = 0.f;
#pragma unroll 8
  for (int i = 0; i < D_ / 4; ++i) {
    v4f xv = row[i], qv = q4[i];
    s += xv.x * qv.x + xv.y * qv.y + xv.z * qv.z + xv.w * qv.w;
  }
  float uv = CLIPV * tanhf(s * (1.0f / 16.0f));      // scale2 = 1/sqrt(256)
  if (mask[(size_t)b * N_ + n]) uv = NEGV;
  u[(size_t)b * N_ + n] = uv;
}

// ---- log_softmax over n, entropy, first-index argmax, outputs, mask upd ----
__global__ void k_select(const float* __restrict__ u, int* __restrict__ mask,
                         int* __restrict__ sel, float* __restrict__ pi,
                         float* __restrict__ ent, int step) {
  __shared__ float rv[256];
  __shared__ int   ri[256];
  int b = blockIdx.x, t = threadIdx.x;
  const float* ub = u + (size_t)b * N_;
  float vals[8];
  float mv = -3.0e38f; int mi = N_;
#pragma unroll
  for (int j = 0; j < 8; ++j) {
    int n = t + j * 256;
    float v = ub[n];
    vals[j] = v;
    if (v > mv || (v == mv && n < mi)) { mv = v; mi = n; }
  }
  rv[t] = mv; ri[t] = mi; __syncthreads();
  for (int s = 128; s > 0; s >>= 1) {
    if (t < s) {
      float ov = rv[t + s]; int oi = ri[t + s];
      if (ov > rv[t] || (ov == rv[t] && oi < ri[t])) { rv[t] = ov; ri[t] = oi; }
    }
    __syncthreads();
  }
  mv = rv[0]; mi = ri[0]; __syncthreads();
  float z = 0.f;
#pragma unroll
  for (int j = 0; j < 8; ++j) z += expf(vals[j] - mv);
  rv[t] = z; __syncthreads();
  for (int s = 128; s > 0; s >>= 1) { if (t < s) rv[t] += rv[t + s]; __syncthreads(); }
  float lse = logf(rv[0]); __syncthreads();
  float e = 0.f;
#pragma unroll
  for (int j = 0; j < 8; ++j) {
    float lp = vals[j] - mv - lse;
    float p  = expf(lp);
    if (p > 0.f) e += p * lp;
  }
  rv[t] = e; __syncthreads();
  for (int s = 128; s > 0; s >>= 1) { if (t < s) rv[t] += rv[t + s]; __syncthreads(); }
  if (t == 0) {
    sel[b * S_ + step] = mi;
    pi [b * S_ + step] = expf(-lse);      // p[argmax] = exp(u_max - m - lse) = exp(-lse)
    ent[b] += -rv[0];
    mask[(size_t)b * N_ + mi] = 1;
  }
}

// ---------------------------------------------------------------------------
extern "C" void kernel_launch(void* const* d_in, const int* in_sizes, int n_in,
                              void* d_out, int out_size, void* d_ws, size_t ws_size,
                              hipStream_t stream) {
  (void)in_sizes; (void)n_in; (void)out_size; (void)ws_size;
  const float* X    = (const float*)d_in[0];
  const float* Wqkv = (const float*)d_in[1];
  const float* bqkv = (const float*)d_in[2];
  const float* Wout = (const float*)d_in[3];
  const float* bout = (const float*)d_in[4];
  const float* Wsq  = (const float*)d_in[5];
  const float* Wsk  = (const float*)d_in[6];
  const float* Wq = Wqkv;
  const float* Wk = Wqkv + (size_t)D_ * D_;
  const float* Wv = Wqkv + (size_t)2 * D_ * D_;
  const float* bq = bqkv;
  const float* bv = bqkv + 2 * D_;

  float* ws = (float*)d_ws;
  float* hpart = ws;                         // 64*16*256          = 262144
  float* hmean = ws + 262144;                // 16384
  float* q     = ws + 278528;                // 16384
  float* q2pad = ws + 294912;                // B*16*D             = 262144
  float* smha  = ws + 557056;                // B*H*N              = 1048576
  float* apad  = ws + 1605632;               // B*16*N             = 2097152
  float* yv    = ws + 3702784;               // 131072
  float* qs2   = ws + 3833856;               // 16384
  float* uv    = ws + 3850240;               // 131072
  int*   mask  = (int*)(ws + 3981312);       // 131072 ints (total ~16.5 MB)

  int*   sel = (int*)d_out;                  // (B,S) int32
  float* pi  = (float*)d_out + B_ * S_;      // (B,S) f32
  float* ent = (float*)d_out + 2 * B_ * S_;  // (B,)  f32

  k_init<<<2048, 256, 0, stream>>>(mask, ent, q2pad, apad);
  k_mean_partial<<<dim3(B_, 16), 256, 0, stream>>>(X, hpart);
  k_mean_final<<<B_, 256, 0, stream>>>(hpart, hmean);
  k_q<<<B_, 256, 0, stream>>>(hmean, Wq, bq, q);
  k_q2<<<dim3(B_, H_), 256, 0, stream>>>(q, Wk, q2pad);
  k_smha<<<dim3(B_, N_ / 128), 256, 0, stream>>>(X, q2pad, smha);
  for (int s = 0; s < S_; ++s) {
    k_softmax<<<dim3(B_, H_), 256, 0, stream>>>(smha, mask, apad);
    k_y<<<dim3(B_, D_ / 16), 256, 0, stream>>>(apad, X, yv);
    k_chain<<<B_, 256, 0, stream>>>(yv, Wv, bv, Wout, bout, Wsq, Wsk, qs2);
    k_u<<<dim3(B_, N_ / 256), 256, 0, stream>>>(X, qs2, mask, uv);
    k_select<<<B_, 256, 0, stream>>>(uv, mask, sel, pi, ent, s);
  }
}